// gpn_layer_14809047236925
// MI455X (gfx1250) — compile-verified
//
#include <hip/hip_runtime.h>
#include <hip/hip_bf16.h>
#include <math.h>
#include <float.h>

typedef __attribute__((ext_vector_type(2))) float v2f;
typedef __attribute__((ext_vector_type(8))) float v8f;

#define NB    16      // batch
#define NN    37      // N objects per subgraph
#define MM    64      // M_SUB
#define LL    1024    // GCN_DIM
#define HID   512
#define GB    2048    // 2*NB*MM
#define RO_W  2048    // read_out width (2*LL)

// ---- output layout (flat, in reference return order) ----
#define OUT_LOSS   0
#define OUT_SCORE  1                       // 2048
#define OUT_ATT    (1 + GB)                // 16*37*1024
#define OUT_FC     (OUT_ATT + NB*NN*LL)    // 16*2048
#define OUT_MASK   (OUT_FC + NB*RO_W)      // 16*37

// =====================================================================
// K1: pooling einsum + max/mean readout via V_WMMA_F32_16X16X4_F32.
// One block per g (2048 blocks, 256 thr = 8 waves).
//   clean(37x1024) = P(37x37) @ A(37x1024), A = gathered att rows.
// P padded to 48x40 (zeros) in LDS; P fragments preloaded to registers
// (invariant across column chunks). A staged transposed in 128-col
// chunks; per K-step: 1 ds_load_b64 (B frag) + 3 WMMAs (M-tiles).
// Row-max / row-sum reduced inside C fragments (clean never stored).
// =====================================================================
#define PP   40     // padded K pitch for P (even -> b64 aligned)
#define ATP  42     // pitch for transposed A chunk (even)
#define CHW  128    // columns per chunk (8 chunks of 1024)

__global__ __launch_bounds__(256)
void k_pool_wmma(const int* __restrict__ objInd,    // (16,2,64,37)
                 const float* __restrict__ pool,    // (16,2,64,37,37)
                 const float* __restrict__ att,     // (16,37,1024)
                 const float* __restrict__ masks,   // (16,2,64,37)
                 float* __restrict__ ro)            // (2048,2048)
{
    __shared__ float Psh[48 * PP];      // P padded 48x40
    __shared__ float At[CHW * ATP];     // A chunk transposed: At[l][j]
    __shared__ int   objSh[NN];
    __shared__ float denomSh;

    const int g   = blockIdx.x;
    const int s   = (g >= NB * MM) ? 1 : 0;
    const int idx = g - s * NB * MM;
    const int bi  = idx / MM;
    const int m   = idx % MM;
    const int tid = threadIdx.x;
    const int lane = tid & 31;
    const int li   = lane & 15;
    const int hi   = lane >> 4;
    const int w    = tid >> 5;          // wave 0..7 -> N-tile within chunk

    const size_t poolBase = ((size_t)(bi * 2 + s) * MM + m) * (NN * NN);
    const size_t objBase  = ((size_t)(bi * 2 + s) * MM + m) * NN;

    // stage padded P and indices
    for (int t = tid; t < 48 * PP; t += 256) {
        const int i = t / PP, k = t % PP;
        Psh[t] = (i < NN && k < NN) ? pool[poolBase + i * NN + k] : 0.f;
    }
    if (tid < NN) objSh[tid] = objInd[objBase + tid];
    if (tid == 0) {
        float d = 0.f;
        for (int n = 0; n < NN; ++n) d += masks[objBase + n];
        denomSh = d;
    }
    __syncthreads();

    const float inv = 1.f / denomSh;

    // preload all P fragments (A operand), invariant across chunks:
    // A frag 16x4 f32: lane=m%16, vgpr0/1 = K+2*hi, K+2*hi+1
    v2f pf[3][10];
#pragma unroll
    for (int tM = 0; tM < 3; ++tM)
#pragma unroll
        for (int ks = 0; ks < 10; ++ks)
            pf[tM][ks] = *(const v2f*)&Psh[(tM * 16 + li) * PP + ks * 4 + 2 * hi];

    for (int lbase = 0; lbase < LL; lbase += CHW) {
        // stage A chunk transposed: At[lc][j] = att[bi, obj[j], lbase+lc]
#pragma unroll
        for (int p = 0; p < 20; ++p) {
            const int j  = (tid >> 7) + 2 * p;      // 0..39
            const int lc = tid & 127;
            const float v = (j < NN)
                ? att[((size_t)bi * NN + objSh[j]) * LL + lbase + lc] : 0.f;
            At[lc * ATP + j] = v;
        }
        __syncthreads();

        v8f acc0 = {}, acc1 = {}, acc2 = {};
#pragma unroll
        for (int ks = 0; ks < 10; ++ks) {
            const v2f bV = *(const v2f*)&At[(w * 16 + li) * ATP + ks * 4 + 2 * hi];
            acc0 = __builtin_amdgcn_wmma_f32_16x16x4_f32(false, pf[0][ks], false, bV,
                                                         (short)0, acc0, false, false);
            acc1 = __builtin_amdgcn_wmma_f32_16x16x4_f32(false, pf[1][ks], false, bV,
                                                         (short)0, acc1, false, false);
            acc2 = __builtin_amdgcn_wmma_f32_16x16x4_f32(false, pf[2][ks], false, bV,
                                                         (short)0, acc2, false, false);
        }

        // reduce over rows inside fragments.
        // C layout: vgpr r, lanes0-15 -> M=r; lanes16-31 -> M=r+8 (per tile).
        // tile0 rows 0-15, tile1 16-31 all valid; tile2 rows 32-47: only
        // 32..36 valid for max (hi==0 && r<=4); sums of pad rows are 0.
        float sm = 0.f, mx = -FLT_MAX;
#pragma unroll
        for (int r = 0; r < 8; ++r) {
            sm += acc0[r] + acc1[r] + acc2[r];
            mx = fmaxf(mx, fmaxf(acc0[r], acc1[r]));
        }
        float mx2 = -FLT_MAX;
#pragma unroll
        for (int r = 0; r < 5; ++r) mx2 = fmaxf(mx2, acc2[r]);
        mx2 = (hi == 0) ? mx2 : -FLT_MAX;
        mx = fmaxf(mx, mx2);

        sm += __shfl_xor(sm, 16, 32);
        mx = fmaxf(mx, __shfl_xor(mx, 16, 32));

        if (lane < 16) {
            const int col = lbase + w * 16 + li;
            ro[(size_t)g * RO_W + col]      = mx;        // max_feat
            ro[(size_t)g * RO_W + LL + col] = sm * inv;  // mean_feat
        }
        __syncthreads();   // protect At before next chunk overwrites
    }
}

// =====================================================================
// K2: h = relu(read_out @ W1 + b1) via V_WMMA_F32_16X16X4_F32.
// Block = 256 thr (8 waves). 64x64 C tile / block, BK=64 in LDS.
// Wave w: tileM = w%4, tileN = {2*(w/4), 2*(w/4)+1}.
// =====================================================================
#define LPITCH 66
__global__ __launch_bounds__(256)
void k_gemm_h(const float* __restrict__ X,    // (2048,2048) read_out
              const float* __restrict__ W1,   // (2048,512)
              const float* __restrict__ b1,   // (512)
              float* __restrict__ H)          // (2048,512)
{
    __shared__ float Xs[64 * LPITCH];
    __shared__ float Wt[64 * LPITCH];   // W1 chunk transposed: Wt[n][k]

    const int lane = threadIdx.x & 31;
    const int li   = lane & 15;
    const int hi   = lane >> 4;
    const int wave = threadIdx.x >> 5;
    const int tM   = wave & 3;
    const int tN0  = (wave >> 2) * 2;

    const int rowBase = blockIdx.x * 64;
    const int colBase = blockIdx.y * 64;

    v8f acc0 = {}, acc1 = {};

    for (int kb = 0; kb < RO_W; kb += 64) {
        // stage X tile (64x64)
#pragma unroll
        for (int p = 0; p < 4; ++p) {
            const int r = (threadIdx.x >> 4) + p * 16;
            const int c = (threadIdx.x & 15) * 4;
            const float4 v = *(const float4*)&X[(size_t)(rowBase + r) * RO_W + kb + c];
            Xs[r * LPITCH + c + 0] = v.x;
            Xs[r * LPITCH + c + 1] = v.y;
            Xs[r * LPITCH + c + 2] = v.z;
            Xs[r * LPITCH + c + 3] = v.w;
        }
        // stage W1 chunk transposed (Wt[n][k])
#pragma unroll
        for (int p = 0; p < 16; ++p) {
            const int kk = (threadIdx.x >> 6) + p * 4;
            const int n  = threadIdx.x & 63;
            Wt[n * LPITCH + kk] = W1[(size_t)(kb + kk) * HID + colBase + n];
        }
        __syncthreads();

#pragma unroll
        for (int kk = 0; kk < 64; kk += 4) {
            const v2f a  = *(const v2f*)&Xs[(tM * 16 + li) * LPITCH + kk + 2 * hi];
            const v2f bA = *(const v2f*)&Wt[(tN0 * 16 + li) * LPITCH + kk + 2 * hi];
            const v2f bB = *(const v2f*)&Wt[((tN0 + 1) * 16 + li) * LPITCH + kk + 2 * hi];
            acc0 = __builtin_amdgcn_wmma_f32_16x16x4_f32(false, a, false, bA,
                                                         (short)0, acc0, false, false);
            acc1 = __builtin_amdgcn_wmma_f32_16x16x4_f32(false, a, false, bB,
                                                         (short)0, acc1, false, false);
        }
        __syncthreads();
    }

    // C/D layout: VGPR r -> M=r (lanes 0-15) / M=r+8 (lanes 16-31), N=lane%16
#pragma unroll
    for (int r = 0; r < 8; ++r) {
        const int row = rowBase + tM * 16 + r + 8 * hi;
        const int c0  = colBase + tN0 * 16 + li;
        const int c1  = c0 + 16;
        H[(size_t)row * HID + c0] = fmaxf(acc0[r] + b1[c0], 0.f);
        H[(size_t)row * HID + c1] = fmaxf(acc1[r] + b1[c1], 0.f);
    }
}

// =====================================================================
// K3: scores = sigmoid(h @ W2 + b2). One wave per row.
// =====================================================================
__global__ __launch_bounds__(256)
void k_score(const float* __restrict__ H, const float* __restrict__ W2,
             const float* __restrict__ b2, float* __restrict__ out)
{
    const int lane = threadIdx.x & 31;
    const int row  = blockIdx.x * 8 + (threadIdx.x >> 5);
    float acc = 0.f;
    for (int k = lane; k < HID; k += 32)
        acc = fmaf(H[(size_t)row * HID + k], W2[k], acc);
#pragma unroll
    for (int off = 16; off > 0; off >>= 1)
        acc += __shfl_xor(acc, off, 32);
    if (lane == 0)
        out[OUT_SCORE + row] = 1.f / (1.f + expf(-(acc + b2[0])));
}

// =====================================================================
// K4: BCE loss reduce -> out[0]
// =====================================================================
__global__ __launch_bounds__(256)
void k_loss(float* __restrict__ out)
{
    __shared__ float red[256];
    float local = 0.f;
    for (int g = threadIdx.x; g < GB; g += 256) {
        const float p  = out[OUT_SCORE + g];
        const float t  = (g < GB / 2) ? 1.f : 0.f;
        const float lp  = fmaxf(logf(p), -100.f);
        const float l1p = fmaxf(log1pf(-p), -100.f);
        local += -(t * lp + (1.f - t) * l1p);
    }
    red[threadIdx.x] = local;
    __syncthreads();
    for (int off = 128; off > 0; off >>= 1) {
        if (threadIdx.x < off) red[threadIdx.x] += red[threadIdx.x + off];
        __syncthreads();
    }
    if (threadIdx.x == 0) out[OUT_LOSS] = red[0] / (float)GB;
}

// =====================================================================
// K5: per-batch argmax over positive scores (first index wins ties)
// =====================================================================
__global__ __launch_bounds__(64)
void k_argmax(const float* __restrict__ out, int* __restrict__ ind)
{
    __shared__ float sv[64];
    __shared__ int   si[64];
    const int tid = threadIdx.x;
    sv[tid] = out[OUT_SCORE + blockIdx.x * MM + tid];
    si[tid] = tid;
    __syncthreads();
    for (int off = 32; off > 0; off >>= 1) {
        if (tid < off) {
            const float a = sv[tid], b = sv[tid + off];
            if (b > a || (b == a && si[tid + off] < si[tid])) {
                sv[tid] = b; si[tid] = si[tid + off];
            }
        }
        __syncthreads();
    }
    if (tid == 0) ind[blockIdx.x] = si[0];
}

// =====================================================================
// K6: t = sub_read_out @ Wp1 + bp1   (16 x 512)
// =====================================================================
__global__ __launch_bounds__(256)
void k_fc1(const float* __restrict__ ro, const int* __restrict__ ind,
           const float* __restrict__ Wp1, const float* __restrict__ bp1,
           float* __restrict__ tbuf)
{
    const int idx = blockIdx.x * 256 + threadIdx.x;   // < 8192
    const int bi  = idx >> 9;
    const int hc  = idx & 511;
    const float* row = ro + (size_t)(bi * MM + ind[bi]) * RO_W;
    float acc = bp1[hc];
    for (int k = 0; k < RO_W; ++k)
        acc = fmaf(row[k], Wp1[(size_t)k * HID + hc], acc);
    tbuf[idx] = acc;
}

// =====================================================================
// K7: fc_out = t @ Wp2 + bp2  (16 x 2048)
// =====================================================================
__global__ __launch_bounds__(256)
void k_fc2(const float* __restrict__ tbuf, const float* __restrict__ Wp2,
           const float* __restrict__ bp2, float* __restrict__ out)
{
    const int idx = blockIdx.x * 256 + threadIdx.x;   // < 32768
    const int bi  = idx >> 11;
    const int c   = idx & 2047;
    float acc = bp2[c];
    for (int k = 0; k < HID; ++k)
        acc = fmaf(tbuf[bi * HID + k], Wp2[(size_t)k * RO_W + c], acc);
    out[OUT_FC + idx] = acc;
}

// =====================================================================
// K8: gathers (att_feats_out, att_masks_out)
// =====================================================================
__global__ __launch_bounds__(256)
void k_gather(const float* __restrict__ att, const int* __restrict__ objInd,
              const float* __restrict__ masks, const int* __restrict__ ind,
              float* __restrict__ out)
{
    const int idx = blockIdx.x * 256 + threadIdx.x;
    if (idx >= NB * NN * LL) return;
    const int bi   = idx / (NN * LL);
    const int rest = idx % (NN * LL);
    const int n    = rest / LL;
    const int l    = rest % LL;
    const int gi   = ind[bi];
    const int o    = objInd[((bi * 2 + 0) * MM + gi) * NN + n];
    out[OUT_ATT + idx] = att[((size_t)bi * NN + o) * LL + l];
    if (l == 0)
        out[OUT_MASK + bi * NN + n] = masks[((bi * 2 + 0) * MM + gi) * NN + n];
}

// =====================================================================
extern "C" void kernel_launch(void* const* d_in, const int* in_sizes, int n_in,
                              void* d_out, int out_size, void* d_ws, size_t ws_size,
                              hipStream_t stream)
{
    const int*   objInd = (const int*)  d_in[4];
    const float* pool   = (const float*)d_in[7];
    const float* att    = (const float*)d_in[8];
    const float* masks  = (const float*)d_in[11];
    const float* W1     = (const float*)d_in[12];
    const float* b1     = (const float*)d_in[13];
    const float* W2     = (const float*)d_in[14];
    const float* b2     = (const float*)d_in[15];
    const float* Wp1    = (const float*)d_in[16];
    const float* bp1    = (const float*)d_in[17];
    const float* Wp2    = (const float*)d_in[18];
    const float* bp2    = (const float*)d_in[19];

    float* out  = (float*)d_out;
    float* ro   = (float*)d_ws;                         // 2048*2048
    float* H    = ro + (size_t)GB * RO_W;               // 2048*512
    float* tbuf = H + (size_t)GB * HID;                 // 16*512
    int*   ind  = (int*)(tbuf + NB * HID);              // 16

    k_pool_wmma<<<GB, 256, 0, stream>>>(objInd, pool, att, masks, ro);
    k_gemm_h<<<dim3(GB / 64, HID / 64), 256, 0, stream>>>(ro, W1, b1, H);
    k_score<<<GB / 8, 256, 0, stream>>>(H, W2, b2, out);
    k_loss<<<1, 256, 0, stream>>>(out);
    k_argmax<<<NB, 64, 0, stream>>>(out, ind);
    k_fc1<<<(NB * HID) / 256, 256, 0, stream>>>(ro, ind, Wp1, bp1, tbuf);
    k_fc2<<<(NB * RO_W) / 256, 256, 0, stream>>>(tbuf, Wp2, bp2, out);
    k_gather<<<(NB * NN * LL + 255) / 256, 256, 0, stream>>>(att, objInd, masks, ind, out);
}